// GraphSAGE_20736102105621
// MI455X (gfx1250) — compile-verified
//
#include <hip/hip_runtime.h>
#include <stdint.h>

#define N_NODES 100000
#define N_EDGES 1600000
#define IN_CH   128
#define HID     128
#define OUT_CH  64

typedef __attribute__((ext_vector_type(2))) float v2f;
typedef __attribute__((ext_vector_type(8))) float v8f;
typedef __attribute__((ext_vector_type(4))) unsigned int u32x4;
typedef __attribute__((ext_vector_type(8))) int          i32x8;
typedef __attribute__((ext_vector_type(4))) int          i32x4;

// ---------------------------------------------------------------------------
// Degree count: deg[dst] += 1 per edge (f32 atomics, resolve in L2).
// ---------------------------------------------------------------------------
__global__ void sage_deg_kernel(const int* __restrict__ dst,
                                float* __restrict__ deg) {
    int e = blockIdx.x * blockDim.x + threadIdx.x;
    if (e < N_EDGES) atomicAdd(&deg[dst[e]], 1.0f);
}

// deg -> inv_deg in place: inv = deg>0 ? 1/deg : 0
__global__ void sage_invdeg_kernel(float* __restrict__ deg) {
    int i = blockIdx.x * blockDim.x + threadIdx.x;
    if (i < N_NODES) {
        float d = deg[i];
        deg[i] = (d > 0.0f) ? (1.0f / d) : 0.0f;
    }
}

// ---------------------------------------------------------------------------
// Scatter-add of 128-channel features along edges: agg[dst] += feat[src].
// 32 threads per edge, 4 channels (one float4) per thread -> coalesced 512B
// gather per edge; atomic adds land in the 192MB L2 (agg is only 51MB).
// ---------------------------------------------------------------------------
__global__ void sage_scatter_kernel(const float* __restrict__ feat,
                                    const int*   __restrict__ src,
                                    const int*   __restrict__ dst,
                                    float*       __restrict__ agg) {
    long long gid = (long long)blockIdx.x * blockDim.x + threadIdx.x;
    int e  = (int)(gid >> 5);
    int cg = (int)(gid & 31);
    if (e >= N_EDGES) return;
    int s = src[e];
    int d = dst[e];
    const float4 v = *(const float4*)(feat + (long long)s * IN_CH + cg * 4);
    float* o = agg + (long long)d * IN_CH + cg * 4;
    atomicAdd(o + 0, v.x);
    atomicAdd(o + 1, v.y);
    atomicAdd(o + 2, v.z);
    atomicAdd(o + 3, v.w);
}

// ---------------------------------------------------------------------------
// TDM: 1-D contiguous DMA of `nelem` f32 from global to LDS (6-arg builtin
// form: g0, g1, g2, g3, g_extra, cpol). D# packing per CDNA5 ISA 8.3/8.4:
//   g0 = { count=1 | flags=0, lds_addr, gaddr[31:0], gaddr[56:32] | type=2 }
//   g1 = { data_size=4B, tensor_dim0=nelem, tensor_dim1=1, tile_dim0=nelem,
//          tile_dim1=1, tensor_dim0_stride=nelem }
// Groups 2/3 zero (tensor <= 2D). Tracked by TENSORcnt.
// ---------------------------------------------------------------------------
__device__ __forceinline__ void tdm_load_1d_f32(uint32_t lds_off_bytes,
                                                const float* gsrc,
                                                uint32_t nelem) {
    uint64_t ga = (uint64_t)(uintptr_t)gsrc;
    u32x4 g0;
    g0[0] = 1u;                                       // count=1
    g0[1] = lds_off_bytes;                            // lds_addr
    g0[2] = (uint32_t)(ga & 0xffffffffu);             // global_addr[31:0]
    g0[3] = (uint32_t)((ga >> 32) & 0x01ffffffu)      // global_addr[56:32]
          | (2u << 30);                               // type=2 ("image")
    i32x8 g1;
    g1[0] = (2 << 16);                                // data_size: 4 bytes
    g1[1] = (int)((nelem & 0xffffu) << 16);           // tensor_dim0[15:0]
    g1[2] = (int)((nelem >> 16) & 0xffffu)            // tensor_dim0[31:16]
          | (1 << 16);                                // tensor_dim1 = 1
    g1[3] = (int)((nelem & 0xffffu) << 16);           // tile_dim0 (<=65535)
    g1[4] = 1;                                        // tile_dim1 = 1
    g1[5] = (int)nelem;                               // tensor_dim0_stride lo
    g1[6] = 0;
    g1[7] = 0;
    i32x4 z4 = {};
    i32x8 z8 = {};
    __builtin_amdgcn_tensor_load_to_lds(g0, g1, z4, z4, z8, 0);
}

// Low 32 bits of a generic pointer to LDS == LDS byte offset (aperture rule).
__device__ __forceinline__ uint32_t lds_offset_of(const void* p) {
    return (uint32_t)(uintptr_t)p;
}

// ---------------------------------------------------------------------------
// Fused SAGE linear: out = act( (agg*inv_deg) @ Wl + bias + xin @ Wr )
//   agg,xin: [N,128] f32   Wl,Wr: [128,NCOLS] row-major   out: [N,NCOLS]
// One wave per 16x16 output tile, V_WMMA_F32_16X16X4_F32 over K=128.
// The 16-row A panels (agg rows, x rows: contiguous 8KB each) are DMA'd
// once per block into LDS via TDM; each wave then reads its A fragments
// from LDS while B (weight) fragments stream from global.
//
// Fragment layouts (ISA 7.12.2, wave32):
//   A 16x4:  lanes 0-15 hold K=0(v0),K=1(v1); lanes 16-31 hold K=2(v0),K=3(v1)
//   B 4x16:  v0 = rows K=0 (lanes 0-15) / K=2 (lanes 16-31); v1 = K=1 / K=3
//   D 16x16: VGPR i -> row M=i (lanes 0-15) / M=8+i (lanes 16-31), N=lane%16
// ---------------------------------------------------------------------------
template <int NCOLS, bool RELU>
__global__ void sage_linear_wmma_kernel(const float* __restrict__ agg,
                                        const float* __restrict__ xin,
                                        const float* __restrict__ invdeg,
                                        const float* __restrict__ Wl,
                                        const float* __restrict__ Wr,
                                        const float* __restrict__ bias,
                                        float*       __restrict__ out) {
    __shared__ float lds_a[2 * 16 * IN_CH];   // [0..2047]=agg rows, [2048..]=x rows

    const int lane = threadIdx.x & 31;
    const int wave = threadIdx.x >> 5;
    const int half = lane >> 4;   // 0: lanes 0-15, 1: lanes 16-31
    const int l    = lane & 15;
    const int row0 = blockIdx.x * 16;
    const int col0 = wave * 16;

    // Wave 0 DMAs both 16x128 A panels into LDS, waits TENSORcnt, then the
    // workgroup barrier releases everyone.
    if (wave == 0) {
        const uint32_t base = lds_offset_of(lds_a);
        tdm_load_1d_f32(base,                          agg + (long long)row0 * IN_CH, 16 * IN_CH);
        tdm_load_1d_f32(base + 16 * IN_CH * sizeof(float),
                                                       xin + (long long)row0 * IN_CH, 16 * IN_CH);
        __builtin_amdgcn_s_wait_tensorcnt(0);
    }
    __syncthreads();

    const int   rowA  = row0 + l;
    const float scale = invdeg[rowA];
    const float* aggp = lds_a + l * IN_CH;             // row l of agg panel
    const float* xp   = lds_a + 16 * IN_CH + l * IN_CH;
    const float* wlp  = Wl + col0 + l;
    const float* wrp  = Wr + col0 + l;

    v8f acc = {};
#pragma unroll
    for (int k = 0; k < IN_CH; k += 4) {
        const int ka = k + 2 * half;            // this lane-half's K pair
        v2f a1, a2, bl, br;
        a1.x = aggp[ka] * scale;  a1.y = aggp[ka + 1] * scale;
        a2.x = xp[ka];            a2.y = xp[ka + 1];
        bl.x = wlp[(long long)ka * NCOLS];  bl.y = wlp[(long long)(ka + 1) * NCOLS];
        br.x = wrp[(long long)ka * NCOLS];  br.y = wrp[(long long)(ka + 1) * NCOLS];
        acc = __builtin_amdgcn_wmma_f32_16x16x4_f32(
            false, a1, false, bl, (short)0, acc, false, false);
        acc = __builtin_amdgcn_wmma_f32_16x16x4_f32(
            false, a2, false, br, (short)0, acc, false, false);
    }

    const float b = bias[col0 + l];
    float* op = out + (long long)(row0 + 8 * half) * NCOLS + col0 + l;
#pragma unroll
    for (int i = 0; i < 8; ++i) {
        float v = acc[i] + b;
        if (RELU) v = fmaxf(v, 0.0f);
        op[(long long)i * NCOLS] = v;
    }
}

// ---------------------------------------------------------------------------
// Launch: deg -> inv_deg -> scatter(x) -> linear1(+relu) -> scatter(h) ->
//         linear2. All on `stream`, graph-capture safe.
// ---------------------------------------------------------------------------
extern "C" void kernel_launch(void* const* d_in, const int* in_sizes, int n_in,
                              void* d_out, int out_size, void* d_ws, size_t ws_size,
                              hipStream_t stream) {
    const float* x   = (const float*)d_in[0];
    const int*   ei  = (const int*)d_in[1];
    const float* W1l = (const float*)d_in[2];
    const float* b1  = (const float*)d_in[3];
    const float* W1r = (const float*)d_in[4];
    const float* W2l = (const float*)d_in[5];
    const float* b2  = (const float*)d_in[6];
    const float* W2r = (const float*)d_in[7];
    const int* src = ei;
    const int* dst = ei + N_EDGES;
    float* out = (float*)d_out;

    // Workspace layout (floats): inv_deg[N] | agg[N*128] | h[N*128]
    float* invdeg = (float*)d_ws;
    float* agg    = invdeg + N_NODES;
    float* h      = agg + (size_t)N_NODES * IN_CH;

    const int scatter_blocks = (int)(((long long)N_EDGES * 32) / 256);

    // Zero deg + agg every call (no cross-call state).
    (void)hipMemsetAsync(invdeg, 0, sizeof(float) * (size_t)N_NODES * (1 + IN_CH), stream);

    sage_deg_kernel<<<(N_EDGES + 255) / 256, 256, 0, stream>>>(dst, invdeg);
    sage_invdeg_kernel<<<(N_NODES + 255) / 256, 256, 0, stream>>>(invdeg);

    // Layer 1
    sage_scatter_kernel<<<scatter_blocks, 256, 0, stream>>>(x, src, dst, agg);
    sage_linear_wmma_kernel<HID, true>
        <<<N_NODES / 16, (HID / 16) * 32, 0, stream>>>(agg, x, invdeg, W1l, W1r, b1, h);

    // Layer 2 (reuse agg buffer, re-zeroed)
    (void)hipMemsetAsync(agg, 0, sizeof(float) * (size_t)N_NODES * IN_CH, stream);
    sage_scatter_kernel<<<scatter_blocks, 256, 0, stream>>>(h, src, dst, agg);
    sage_linear_wmma_kernel<OUT_CH, false>
        <<<N_NODES / 16, (OUT_CH / 16) * 32, 0, stream>>>(agg, h, invdeg, W2l, W2r, b2, out);
}